// GNNEncoder_28561532519143
// MI455X (gfx1250) — compile-verified
//
#include <hip/hip_runtime.h>

typedef __attribute__((ext_vector_type(2))) float v2f;
typedef __attribute__((ext_vector_type(4))) float v4f;
typedef __attribute__((ext_vector_type(8))) float v8f;

// ---------------------------------------------------------------------------
// GNN encoder: per-batch graph of N=64 nodes, D=32 in-features, H=64 hidden.
//   A_hat = sym_normalize(adj(pos) + I)
//   h1 = relu((A_hat @ obs) @ W1 + b1)
//   out = relu((A_hat @ h1) @ W2 + b2)
// One workgroup (128 threads = 4 wave32) per batch. All GEMMs on
// V_WMMA_F32_16X16X4_F32 (f32 end-to-end; HBM-bound, ~8us roofline at 23TB/s).
// Streaming traffic (obs in, out) is non-temporal; weights stay L2-resident.
// ---------------------------------------------------------------------------

__device__ __forceinline__ v2f load_a_frag(const float* S, int ld, int rbase, int k, int lane) {
  // A 16x4 f32 fragment: lane&15 -> row, (lane>>4)*2 + {0,1} -> k
  const float* p = S + (rbase + (lane & 15)) * ld + k + ((lane >> 4) << 1);
  return *(const v2f*)p;  // ld even, k multiple of 4 -> 8B aligned
}

__device__ __forceinline__ v2f load_b_frag(const float* S, int ld, int k, int col, int lane) {
  // B 4x16 f32 fragment: lane&15 -> col, (lane>>4)*2 + vgpr -> k (row)
  int kk = k + ((lane >> 4) << 1);
  int c = col + (lane & 15);
  v2f b;
  b.x = S[kk * ld + c];
  b.y = S[(kk + 1) * ld + c];
  return b;
}

__device__ __forceinline__ void store_c_lds(float* S, int ld, int rbase, int col, int lane, v8f c) {
  // C/D 16x16 f32: lane&15 -> col, row = rbase + 8*(lane>>4) + vgpr
  int cc = col + (lane & 15);
  int r0 = rbase + ((lane >> 4) << 3);
#pragma unroll
  for (int v = 0; v < 8; ++v) S[(r0 + v) * ld + cc] = c[v];
}

template <int NT>
__device__ __forceinline__ void wave_gemm(const float* A, int lda, const float* B, int ldb,
                                          int K, int rbase, int lane, v8f acc[NT]) {
  for (int k = 0; k < K; k += 4) {
    v2f a = load_a_frag(A, lda, rbase, k, lane);
#pragma unroll
    for (int t = 0; t < NT; ++t) {
      v2f b = load_b_frag(B, ldb, k, t * 16, lane);
      acc[t] = __builtin_amdgcn_wmma_f32_16x16x4_f32(
          /*neg_a=*/false, a, /*neg_b=*/false, b,
          /*c_mod=*/(short)0, acc[t], /*reuse_a=*/false, /*reuse_b=*/false);
    }
  }
}

__global__ __launch_bounds__(128) void gnn_encoder_kernel(
    const float* __restrict__ obs, const float* __restrict__ W1,
    const float* __restrict__ b1, const float* __restrict__ W2,
    const float* __restrict__ b2, float* __restrict__ out) {
  constexpr int N = 64, D = 32, H = 64;

  __shared__ __align__(16) float Ah[N * N];       // A_hat; later reused for M2 (own rows only)
  __shared__ __align__(16) float obs_s[N * D];
  __shared__ __align__(16) float W1s[D * H];
  __shared__ __align__(16) float W2s[H * H];
  __shared__ __align__(16) float M1s[N * D];
  __shared__ __align__(16) float H1s[N * H];      // H1; later reused as OUT staging
  __shared__ __align__(16) float b1s[H];
  __shared__ __align__(16) float b2s[H];
  __shared__ __align__(16) float posx[N];
  __shared__ __align__(16) float posy[N];
  __shared__ __align__(16) float dinv_s[N];
  __shared__ __align__(16) float psum[128];

  const int b = blockIdx.x;
  const int t = threadIdx.x;
  const int lane = t & 31;
  const int wave = t >> 5;
  const int rbase = wave * 16;

  // ---- stage weights (regular, L2-resident) + obs (non-temporal stream) ----
  {
    const v4f* W1v = (const v4f*)W1;
    v4f* W1sv = (v4f*)W1s;
#pragma unroll
    for (int i = t; i < (D * H) / 4; i += 128) W1sv[i] = W1v[i];
    const v4f* W2v = (const v4f*)W2;
    v4f* W2sv = (v4f*)W2s;
#pragma unroll
    for (int i = t; i < (H * H) / 4; i += 128) W2sv[i] = W2v[i];
    const v4f* ov = (const v4f*)(obs + (size_t)b * (N * D));
    v4f* osv = (v4f*)obs_s;
#pragma unroll
    for (int i = t; i < (N * D) / 4; i += 128) osv[i] = __builtin_nontemporal_load(&ov[i]);
    if (t < H) { b1s[t] = b1[t]; b2s[t] = b2[t]; }
  }
  __syncthreads();

  if (t < N) {
    posx[t] = obs_s[t * D + 2];
    posy[t] = obs_s[t * D + 3];
  }
  __syncthreads();

  // ---- adjacency + row-degree partial sums (thread t: row t>>1, half t&1) ----
  {
    const int r = t >> 1, c0 = (t & 1) * 32;
    const float px = posx[r], py = posy[r];
    float s = 0.f;
#pragma unroll
    for (int j = 0; j < 32; ++j) {
      const int c = c0 + j;
      const float dx = px - posx[c], dy = py - posy[c];
      // dist <= 0.5  <=>  dist^2 <= 0.25 ; diagonal: dist == 0 -> 1 (matches min(adj+I,1))
      const float a = (dx * dx + dy * dy <= 0.25f) ? 1.0f : 0.0f;
      Ah[r * N + c] = a;
      s += a;
    }
    psum[t] = s;
  }
  __syncthreads();
  if (t < N) {
    float deg = psum[t * 2] + psum[t * 2 + 1];
    deg = fmaxf(deg, 1e-8f);
    dinv_s[t] = rsqrtf(deg);  // deg^(-1/2)
  }
  __syncthreads();
  {
    const int r = t >> 1, c0 = (t & 1) * 32;
    const float dr = dinv_s[r];
#pragma unroll
    for (int j = 0; j < 32; ++j) {
      const int c = c0 + j;
      Ah[r * N + c] *= dr * dinv_s[c];
    }
  }
  __syncthreads();

  // ---- M1 = A_hat(64x64) @ obs(64x32) ----
  {
    v8f acc[2] = {};
    wave_gemm<2>(Ah, N, obs_s, D, /*K=*/N, rbase, lane, acc);
    store_c_lds(M1s, D, rbase, 0, lane, acc[0]);
    store_c_lds(M1s, D, rbase, 16, lane, acc[1]);
  }
  __syncthreads();

  // ---- H1 = relu(M1(64x32) @ W1(32x64) + b1) ----
  {
    v8f acc[4] = {};
    wave_gemm<4>(M1s, D, W1s, H, /*K=*/D, rbase, lane, acc);
#pragma unroll
    for (int tc = 0; tc < 4; ++tc) {
      const float bias = b1s[tc * 16 + (lane & 15)];
#pragma unroll
      for (int v = 0; v < 8; ++v) acc[tc][v] = fmaxf(acc[tc][v] + bias, 0.0f);
      store_c_lds(H1s, H, rbase, tc * 16, lane, acc[tc]);
    }
  }
  __syncthreads();  // H1 read as B by all waves

  // ---- M2 = A_hat(64x64) @ H1(64x64); overwrite Ah row block in place ----
  // (wave w reads only Ah rows [16w,16w+16) as its A operand -> no cross-wave hazard)
  {
    v8f acc[4] = {};
    wave_gemm<4>(Ah, N, H1s, H, /*K=*/N, rbase, lane, acc);
#pragma unroll
    for (int tc = 0; tc < 4; ++tc) store_c_lds(Ah, N, rbase, tc * 16, lane, acc[tc]);
  }
  __syncthreads();  // after this, every wave is done reading H1s -> safe to reuse

  // ---- OUT = relu(M2(64x64) @ W2(64x64) + b2); stage in LDS, then b128 NT stores ----
  {
    v8f acc[4] = {};
    wave_gemm<4>(Ah, N, W2s, H, /*K=*/H, rbase, lane, acc);
#pragma unroll
    for (int tc = 0; tc < 4; ++tc) {
      const float bias = b2s[tc * 16 + (lane & 15)];
#pragma unroll
      for (int v = 0; v < 8; ++v) acc[tc][v] = fmaxf(acc[tc][v] + bias, 0.0f);
      store_c_lds(H1s, H, rbase, tc * 16, lane, acc[tc]);  // reuse H1s as OUT staging
    }
  }
  __syncthreads();

  // fully-coalesced 128-bit non-temporal stores of the 16 KB tile
  {
    const v4f* src = (const v4f*)H1s;
    v4f* dst = (v4f*)(out + (size_t)b * (N * H));
#pragma unroll
    for (int i = t; i < (N * H) / 4; i += 128) {
      __builtin_nontemporal_store(src[i], &dst[i]);
    }
  }
}

extern "C" void kernel_launch(void* const* d_in, const int* in_sizes, int n_in,
                              void* d_out, int out_size, void* d_ws, size_t ws_size,
                              hipStream_t stream) {
  const float* obs = (const float*)d_in[0];
  const float* W1 = (const float*)d_in[1];
  const float* b1 = (const float*)d_in[2];
  const float* W2 = (const float*)d_in[3];
  const float* b2 = (const float*)d_in[4];
  float* out = (float*)d_out;

  const int nbatch = in_sizes[0] / (64 * 32);  // B = 8192
  gnn_encoder_kernel<<<dim3(nbatch), dim3(128), 0, stream>>>(obs, W1, b1, W2, b2, out);
}